// Focal_loss_sigmoid3_44040594653648
// MI455X (gfx1250) — compile-verified
//
#include <hip/hip_runtime.h>
#include <hip/hip_bf16.h>

// ---------------- problem constants (from reference) ----------------
#define NCLS 3
#define B_ 4
#define A_ 3
#define D_ 64
#define H_ 96
#define W_ 96
#define K_ 64
#define EPSV 1.0e-4f

#define PLANE (H_ * W_)                 // 9216
#define VOL   ((long)D_ * PLANE)        // 589824
#define NVOX  ((long)B_ * A_ * VOL)     // 7,077,888

// ---------------- pass-2 (pooling) tiling ----------------
#define TD2 8
#define TH2 12
#define DT2 (D_ / TD2)                  // 8
#define HT2 (H_ / TH2)                  // 8
#define ZD (TD2 + 2)                    // 10
#define ZH (TH2 + 2)                    // 14
#define ZW 104                          // padded LDS row pitch (floats); w=0 at slot 4 (16B aligned)
#define NROWS (ZD * ZH)                 // 140 halo rows
#define NQ (NROWS * 24)                 // 3360 b128 quads (96 floats/row)
#define NH2 (NROWS * ZW)                // 14560 floats -> 58.2KB LDS
#define NI2 (TD2 * TH2 * W_)            // 9216 interior voxels
#define NEG_BLOCKS (B_ * A_ * DT2 * HT2) // 768

// ---------------- workspace layout (in floats) ----------------
#define WS_NEG_PART 0                   // NEG_BLOCKS * 2 floats
#define WS_POS_PART 2048                // B_ * 4 floats
#define WS_PMAX     4096                // NVOX floats (~28.3 MB)

// ---------------- CDNA5 async global->LDS copies ----------------
__device__ __forceinline__ void async_ld_b128(void* lds_dst, const float* gsrc) {
#if defined(__gfx1250__)
  unsigned loff = (unsigned)(size_t)lds_dst; // low 32 bits of flat addr == LDS offset
  asm volatile("global_load_async_to_lds_b128 %0, %1, off"
               :
               : "v"(loff), "v"(gsrc)
               : "memory");
#else
  *(float4*)lds_dst = *(const float4*)gsrc;
#endif
}

__device__ __forceinline__ void async_wait0() {
#if defined(__gfx1250__)
  asm volatile("s_wait_asynccnt 0" ::: "memory");
#endif
}

__device__ __forceinline__ float sigmoid_clip(float x) {
  float p = 1.0f / (1.0f + __expf(-x));
  return fminf(fmaxf(p, EPSV), 1.0f - EPSV);
}

// ---------------- pass 1: pmax = clip(max_cls sigmoid(logit)) ----------------
__global__ __launch_bounds__(256) void pmax_kernel(const float* __restrict__ logit,
                                                   float* __restrict__ pmax) {
  long i = (long)blockIdx.x * 256 + threadIdx.x;
  if (i >= NVOX) return;
  long vox = i % VOL;                 // d*PLANE + h*W + w
  int ba = (int)(i / VOL);            // b*A_ + a
  int b = ba / A_, a = ba % A_;
  const float* base = logit + (long)(b * (NCLS * A_) + a) * VOL + vox;
  float pm = -__builtin_inff();
#pragma unroll
  for (int c = 0; c < NCLS; ++c) {
    float x = base[(long)c * A_ * VOL];       // channel = c*A_ + a
    pm = fmaxf(pm, 1.0f / (1.0f + __expf(-x)));
  }
  pm = fminf(fmaxf(pm, EPSV), 1.0f - EPSV);   // clip(max) == max(clip)
  pmax[i] = pm;
}

// ---------------- pass 2: 3x3x3 local-max + negative focal terms ----------------
__global__ __launch_bounds__(256) void neg_kernel(const float* __restrict__ pmax,
                                                  const float* __restrict__ pgt,
                                                  float* __restrict__ ws) {
  __shared__ float s_p[NH2];
  __shared__ float s_red[256];
  const int tid = threadIdx.x;
  const int ba = blockIdx.y;                  // b*A_ + a
  const int dt = blockIdx.x / HT2;
  const int ht = blockIdx.x % HT2;
  const int d0 = dt * TD2, h0 = ht * TH2;
  const float* pm_base = pmax + (long)ba * VOL;
  const float NEGINF = -__builtin_inff();

  // Stage halo tile of pmax into LDS.
  // In-bounds rows: 24 x async b128 per row (96 floats, both sides 16B aligned).
  // OOB rows: -inf via plain ds_store (disjoint from async destinations).
  for (int q = tid; q < NQ; q += 256) {
    int row = q / 24, qi = q - row * 24;
    int hz = row / ZH, hy = row - hz * ZH;
    int d = d0 - 1 + hz, h = h0 - 1 + hy;
    float* dst = &s_p[row * ZW + 4 + qi * 4];
    if ((unsigned)d < (unsigned)D_ && (unsigned)h < (unsigned)H_) {
      async_ld_b128(dst, pm_base + (long)d * PLANE + h * W_ + qi * 4);
    } else {
      *(float4*)dst = make_float4(NEGINF, NEGINF, NEGINF, NEGINF);
    }
  }
  // W-halo columns (global w = -1 and w = 96) are always outside the domain.
  for (int e = tid; e < NROWS * 2; e += 256) {
    int row = e >> 1;
    s_p[row * ZW + ((e & 1) ? 100 : 3)] = NEGINF;
  }
  async_wait0();
  __syncthreads();

  float lneg = 0.0f, cneg = 0.0f;
  const float* pg = pgt + (long)ba * VOL;
  for (int idx = tid; idx < NI2; idx += 256) {
    int iz = idx / (TH2 * W_);
    int r  = idx - iz * (TH2 * W_);
    int iy = r / W_;
    int ix = r - iy * W_;
    int row = (iz + 1) * ZH + (iy + 1);
    int col = ix + 4;
    float center = s_p[row * ZW + col];
    float g = pg[(long)(d0 + iz) * PLANE + (h0 + iy) * W_ + ix];
    if (g == -1.0f && center > EPSV) {
      // branchless 3x3x3 window max; pooled == pmax <=> window max <= center
      float m = NEGINF;
#pragma unroll
      for (int dz = -1; dz <= 1; ++dz) {
#pragma unroll
        for (int dy = -1; dy <= 1; ++dy) {
          const float* rp = &s_p[(row + dz * ZH + dy) * ZW + col];
          m = fmaxf(m, fmaxf(fmaxf(rp[-1], rp[0]), rp[1]));
        }
      }
      if (m <= center) {
        float wn = center * center;             // pmax^ALPHA, ALPHA=2
        lneg += -__logf(1.0f - center) * wn;
        cneg += wn;
      }
    }
  }

  // deterministic block reduction -> per-block partials
  s_red[tid] = lneg; __syncthreads();
  for (int s = 128; s > 0; s >>= 1) { if (tid < s) s_red[tid] += s_red[tid + s]; __syncthreads(); }
  float tl = s_red[0]; __syncthreads();
  s_red[tid] = cneg; __syncthreads();
  for (int s = 128; s > 0; s >>= 1) { if (tid < s) s_red[tid] += s_red[tid + s]; __syncthreads(); }
  if (tid == 0) {
    int blin = ba * (DT2 * HT2) + blockIdx.x;
    ws[WS_NEG_PART + blin * 2 + 0] = tl;
    ws[WS_NEG_PART + blin * 2 + 1] = s_red[0];
  }
}

// ---------------- positive / other-class terms (tiny) ----------------
__global__ __launch_bounds__(64) void pos_kernel(const float* __restrict__ logit,
                                                 const float* __restrict__ pgt,
                                                 const int* __restrict__ coord,
                                                 const float* __restrict__ wcls,
                                                 float* __restrict__ ws) {
  __shared__ float s_keep[NCLS];
  __shared__ float s_val[K_][4];
  const int b = blockIdx.x;
  const int k = threadIdx.x;
  int4 c4 = ((const int4*)coord)[b * K_ + k];
  bool valid = c4.x > -1;
  if (k < NCLS) s_keep[k] = 1.0f;
  __syncthreads();

  int cls_gt = 0;
  long voff = 0;
  if (valid) {
    int a = c4.x, d = c4.y, h = c4.z, w = c4.w;
    voff = (long)d * PLANE + h * W_ + w;
    float g = pgt[(long)(b * A_ + a) * VOL + voff];
    cls_gt = min(max((int)g - 1, 0), NCLS - 1);
    s_keep[cls_gt] = 0.0f;   // benign same-value race across points
  }
  __syncthreads();

  float lp = 0.f, cp = 0.f, lo = 0.f, co = 0.f;
  if (valid) {
    int a = c4.x;
    float p[NCLS];
#pragma unroll
    for (int c = 0; c < NCLS; ++c) {
      float x = logit[(long)(b * (NCLS * A_) + c * A_ + a) * VOL + voff];
      p[c] = sigmoid_clip(x);
    }
    float p_t = p[cls_gt];
    float om = 1.0f - p_t;
    float wpos = om * om * wcls[cls_gt];
    lp = -__logf(p_t) * wpos;
    cp = wpos;
    float gate_t = (p_t > 0.5f) ? 1.0f : 0.0f;
#pragma unroll
    for (int c = 0; c < NCLS; ++c) {
      float po = p[c];
      float woth = fmaxf(po - (p_t - 0.05f), 0.0f)
                 * ((po > 0.5f) ? 1.0f : 0.0f) * gate_t * s_keep[c];
      lo += -__logf(1.0f - po) * woth;
      co += (woth > 0.0f) ? 1.0f : 0.0f;
    }
  }
  s_val[k][0] = lp; s_val[k][1] = cp; s_val[k][2] = lo; s_val[k][3] = co;
  __syncthreads();
  if (k == 0) {
    float a0 = 0, a1 = 0, a2 = 0, a3 = 0;
    for (int i = 0; i < K_; ++i) {
      a0 += s_val[i][0]; a1 += s_val[i][1]; a2 += s_val[i][2]; a3 += s_val[i][3];
    }
    ws[WS_POS_PART + b * 4 + 0] = a0;  // loss_pos
    ws[WS_POS_PART + b * 4 + 1] = a1;  // count_pos
    ws[WS_POS_PART + b * 4 + 2] = a2;  // loss_oth
    ws[WS_POS_PART + b * 4 + 3] = a3;  // count_oth
  }
}

// ---------------- final fixed-order reduction -> 6 outputs ----------------
__global__ __launch_bounds__(256) void reduce_kernel(const float* __restrict__ ws,
                                                     float* __restrict__ out) {
  __shared__ float s_red[256];
  int tid = threadIdx.x;
  float ln = 0.f, cn = 0.f;
  for (int i = tid; i < NEG_BLOCKS; i += 256) {
    ln += ws[WS_NEG_PART + i * 2 + 0];
    cn += ws[WS_NEG_PART + i * 2 + 1];
  }
  s_red[tid] = ln; __syncthreads();
  for (int s = 128; s > 0; s >>= 1) { if (tid < s) s_red[tid] += s_red[tid + s]; __syncthreads(); }
  float loss_neg = s_red[0]; __syncthreads();
  s_red[tid] = cn; __syncthreads();
  for (int s = 128; s > 0; s >>= 1) { if (tid < s) s_red[tid] += s_red[tid + s]; __syncthreads(); }
  if (tid == 0) {
    float count_neg = s_red[0];
    const float* pp = ws + WS_POS_PART;
    float lp = 0, cp = 0, lo = 0, co = 0;
    for (int b = 0; b < B_; ++b) {
      lp += pp[b * 4 + 0]; cp += pp[b * 4 + 1]; lo += pp[b * 4 + 2]; co += pp[b * 4 + 3];
    }
    out[0] = lp;        // sum(loss_pos)
    out[1] = loss_neg;  // loss_neg
    out[2] = lo;        // sum(loss_oth)
    out[3] = cp;        // sum(count_pos)
    out[4] = count_neg; // count_neg
    out[5] = co;        // sum(count_oth)
  }
}

extern "C" void kernel_launch(void* const* d_in, const int* in_sizes, int n_in,
                              void* d_out, int out_size, void* d_ws, size_t ws_size,
                              hipStream_t stream) {
  (void)in_sizes; (void)n_in; (void)out_size; (void)ws_size;
  const float* logit = (const float*)d_in[0];
  const float* pgt   = (const float*)d_in[1];
  const int*   coord = (const int*)d_in[2];
  const float* wcls  = (const float*)d_in[3];
  float* out = (float*)d_out;
  float* ws  = (float*)d_ws;
  float* pmax = ws + WS_PMAX;

  unsigned nblk1 = (unsigned)((NVOX + 255) / 256);   // 27648
  pmax_kernel<<<dim3(nblk1), dim3(256), 0, stream>>>(logit, pmax);
  neg_kernel<<<dim3(DT2 * HT2, B_ * A_), dim3(256), 0, stream>>>(pmax, pgt, ws);
  pos_kernel<<<dim3(B_), dim3(K_), 0, stream>>>(logit, pgt, coord, wcls, ws);
  reduce_kernel<<<dim3(1), dim3(256), 0, stream>>>(ws, out);
}